// KPC_fu_46351287059054
// MI455X (gfx1250) — compile-verified
//
#include <hip/hip_runtime.h>
#include <hip/hip_bf16.h>
#include <math.h>
#include <stdint.h>

typedef float v2f __attribute__((ext_vector_type(2)));
typedef float v8f __attribute__((ext_vector_type(8)));
typedef unsigned int u32x4 __attribute__((ext_vector_type(4)));
typedef int i32x4 __attribute__((ext_vector_type(4)));
typedef int i32x8 __attribute__((ext_vector_type(8)));

#define C6   6
#define MID3 3
#define B2   2
#define NPT  32768
#define LL   2048
#define LPAD (LL + 16)      // 16-col pad so the software-pipeline prefetch stays in-bounds
#define NBLK2 512           // attn kernel blocks: B * (N / 128)

// d_ws float offsets
#define WS_KG   0                     // B*3*L   k_intra
#define WS_VG   (WS_KG + B2*3*LL)     // B*L*3   v_intra
#define WS_KIN  (WS_VG + B2*3*LL)     // B*9     k_inter
#define WS_VIN  (WS_KIN + B2*9)       // B*9     v_inter
#define WS_BNP  (WS_VIN + B2*9)       // NBLK2*12 per-block BN partials
#define WS_BNS  (WS_BNP + NBLK2*12)   // 12      reduced BN stats
// total = 30768 floats = 123072 bytes

#if defined(__has_builtin)
#if __has_builtin(__builtin_amdgcn_tensor_load_to_lds) && __has_builtin(__builtin_amdgcn_s_wait_tensorcnt)
#define HAVE_TDM 1
#endif
#endif

// ---------- block reduction helpers (256 threads) ----------
__device__ __forceinline__ float bsum(float v, float* red) {
  int t = threadIdx.x;
  __syncthreads(); red[t] = v; __syncthreads();
  for (int s = 128; s > 0; s >>= 1) { if (t < s) red[t] += red[t + s]; __syncthreads(); }
  return red[0];
}
__device__ __forceinline__ float bmax(float v, float* red) {
  int t = threadIdx.x;
  __syncthreads(); red[t] = v; __syncthreads();
  for (int s = 128; s > 0; s >>= 1) { if (t < s) red[t] = fmaxf(red[t], red[t + s]); __syncthreads(); }
  return red[0];
}

// ============ Kernel 1: curves preprocessing -> k/v inter & intra ============
__global__ __launch_bounds__(256) void prep_kernel(
    const float* __restrict__ curves, const float* __restrict__ Wa,
    const float* __restrict__ Wav,    const float* __restrict__ Wb,
    const float* __restrict__ Wbv,    const float* __restrict__ Watt,
    const float* __restrict__ Wpl,    const float* __restrict__ bpl,
    const float* __restrict__ Wpn,    const float* __restrict__ bpn,
    float* __restrict__ ws)
{
  __shared__ float att[3][LL];      // 24 KB
  __shared__ float red[256];
  __shared__ float cinter[C6][3];   // curver_inter for this batch
  const int b = blockIdx.x;
  const int tid = threadIdx.x;

  float watt[C6];
  #pragma unroll
  for (int c = 0; c < C6; ++c) watt[c] = Watt[c];

  // line attention logits: att[n][l] = sum_c curves[b,c,n,l] * Watt[c]
  for (int i = tid; i < 3 * LL; i += 256) {
    int n = i / LL, l = i - n * LL;
    float s = 0.f;
    #pragma unroll
    for (int c = 0; c < C6; ++c)
      s += curves[((b * C6 + c) * 3 + n) * LL + l] * watt[c];
    att[n][l] = s;
  }
  __syncthreads();

  // softmax over L (axis=-1) -> curver_inter[c][n]
  for (int n = 0; n < 3; ++n) {
    float pm = -3.4e38f;
    for (int l = tid; l < LL; l += 256) pm = fmaxf(pm, att[n][l]);
    float mx = bmax(pm, red);
    float wsum = 0.f, pc[C6];
    #pragma unroll
    for (int c = 0; c < C6; ++c) pc[c] = 0.f;
    for (int l = tid; l < LL; l += 256) {
      float w = __expf(att[n][l] - mx);
      wsum += w;
      #pragma unroll
      for (int c = 0; c < C6; ++c)
        pc[c] += curves[((b * C6 + c) * 3 + n) * LL + l] * w;
    }
    float den = bsum(wsum, red);
    for (int c = 0; c < C6; ++c) {
      float tot = bsum(pc[c], red);
      if (tid == 0) cinter[c][n] = tot / den;
    }
  }

  // p means: mean_l curves[0,c,k,l]  (p = curves[0] regardless of b)
  float pacc[C6][3];
  #pragma unroll
  for (int c = 0; c < C6; ++c)
    #pragma unroll
    for (int k = 0; k < 3; ++k) pacc[c][k] = 0.f;
  for (int l = tid; l < LL; l += 256)
    #pragma unroll
    for (int c = 0; c < C6; ++c)
      #pragma unroll
      for (int k = 0; k < 3; ++k)
        pacc[c][k] += curves[(c * 3 + k) * LL + l];
  __shared__ float pmean[C6][3];
  for (int c = 0; c < C6; ++c)
    for (int k = 0; k < 3; ++k) {
      float tot = bsum(pacc[c][k], red);
      if (tid == 0) pmean[c][k] = tot * (1.f / (float)LL);
    }
  __syncthreads();

  // tiny: gi, k_inter, v_inter (thread 0)
  if (tid == 0) {
    float gi[C6][3];
    for (int c = 0; c < C6; ++c)
      for (int k = 0; k < 3; ++k) {
        // p_l[c][m] = sum_k Wpl[m,k]*pmean[c][k] + bpl[m]
        float pl = bpl[k];
        for (int kk = 0; kk < 3; ++kk) pl += Wpl[k * 3 + kk] * pmean[c][kk];
        gi[c][k] = cinter[c][k] + pl;
      }
    for (int m = 0; m < MID3; ++m)
      for (int k = 0; k < 3; ++k) {
        float kv = 0.f, vv = 0.f;
        for (int c = 0; c < C6; ++c) {
          kv += Wa[m * C6 + c]  * gi[c][k];
          vv += Wav[m * C6 + c] * gi[c][k];
        }
        ws[WS_KIN + b * 9 + m * 3 + k] = kv;   // k_inter[m][k]
        ws[WS_VIN + b * 9 + k * 3 + m] = vv;   // v_inter[k][m]
      }
  }

  // softmax over the 3-axis + p_n fold + k_intra / v_intra
  float wb[MID3][C6], wbv[MID3][C6], wbar[3], bbar;
  #pragma unroll
  for (int m = 0; m < MID3; ++m)
    #pragma unroll
    for (int c = 0; c < C6; ++c) { wb[m][c] = Wb[m * C6 + c]; wbv[m][c] = Wbv[m * C6 + c]; }
  #pragma unroll
  for (int k = 0; k < 3; ++k)
    wbar[k] = (Wpn[0 * 3 + k] + Wpn[1 * 3 + k] + Wpn[2 * 3 + k]) * (1.f / 3.f);
  bbar = (bpn[0] + bpn[1] + bpn[2]) * (1.f / 3.f);

  for (int l = tid; l < LL; l += 256) {
    float a0 = att[0][l], a1 = att[1][l], a2 = att[2][l];
    float m3 = fmaxf(a0, fmaxf(a1, a2));
    float e0 = __expf(a0 - m3), e1 = __expf(a1 - m3), e2 = __expf(a2 - m3);
    float inv = 1.f / (e0 + e1 + e2);
    float ga[C6];
    #pragma unroll
    for (int c = 0; c < C6; ++c) {
      const float* cc = curves + ((b * C6 + c) * 3) * LL + l;
      float ci = (cc[0] * e0 + cc[LL] * e1 + cc[2 * LL] * e2) * inv;
      const float* pp = curves + (c * 3) * LL + l;   // batch 0
      float pn = wbar[0] * pp[0] + wbar[1] * pp[LL] + wbar[2] * pp[2 * LL] + bbar;
      ga[c] = ci + pn;
    }
    #pragma unroll
    for (int m = 0; m < MID3; ++m) {
      float kv = 0.f, vv = 0.f;
      #pragma unroll
      for (int c = 0; c < C6; ++c) { kv += wb[m][c] * ga[c]; vv += wbv[m][c] * ga[c]; }
      ws[WS_KG + b * 3 * LL + m * LL + l] = kv;   // k_intra[m][l]
      ws[WS_VG + b * 3 * LL + l * 3 + m] = vv;    // v_intra[l][m]
    }
  }
}

// ============ Kernel 2: fused attention (WMMA f32 16x16x4 + TDM staging) ============
__global__ __launch_bounds__(256) void attn_kernel(
    const float* __restrict__ x, const float* __restrict__ Wc,
    const float* __restrict__ Wd, const float* __restrict__ lng,
    const float* __restrict__ lnb, float* ws, float* __restrict__ yout)
{
  __shared__ float klds[2][LPAD][2]; // 33 KB: K-pairs for WMMA B operand, K=3 padded to 4
  __shared__ float vlds[LPAD * 3];   // 24.8 KB (TDM fills first 6144 floats)
  __shared__ float xch[8][16][4];    // per-wave row exchange {f0,f1,f2,denom}
  __shared__ float bred[8][12];      // per-wave BN partials
  const int tid = threadIdx.x;
  const int b  = blockIdx.x >> 8;          // 256 blocks per batch
  const int n0 = (blockIdx.x & 255) * 128; // 128 points per block

  const float* kg = ws + WS_KG + b * 3 * LL;
  const float* vg = ws + WS_VG + b * 3 * LL;

#ifdef HAVE_TDM
  // --- stage v_intra (6144 contiguous f32) into LDS via the Tensor Data Mover ---
  // D# per CDNA5 ISA ch.8: group0 = {count, lds_addr, global_addr, type=2},
  // group1 = {mask=0, data_size=4B, dims}. 1-D tile: tile_dim0 = tensor_dim0 = 6144.
  if (tid < 32) {                      // one wave issues the DMA (EXEC-independent op)
    uint32_t lds_v = (uint32_t)(uintptr_t)&vlds[0];   // LDS aperture: offset = addr[31:0]
    uint64_t ga    = (uint64_t)(uintptr_t)vg;
    u32x4 g0;
    g0[0] = 1u;                                       // count=1, is_restore=0, gather off
    g0[1] = lds_v;                                    // lds_addr
    g0[2] = (uint32_t)(ga & 0xffffffffu);             // global_addr[31:0]
    g0[3] = (uint32_t)((ga >> 32) & 0x01ffffffu) | (2u << 30);  // addr[56:32] | type=2
    i32x8 g1;
    g1[0] = (int)(2u << 16);                          // workgroup_mask=0, data_size=4B
    g1[1] = (int)((6144u & 0xffffu) << 16);           // tensor_dim0[15:0]
    g1[2] = (int)((6144u >> 16) | (1u << 16));        // tensor_dim0[31:16] | tensor_dim1=1
    g1[3] = (int)((6144u & 0xffffu) << 16);           // tensor_dim1[31:16]=0 | tile_dim0=6144
    g1[4] = 1;                                        // tile_dim1=1, tile_dim2=0
    g1[5] = 6144;                                     // tensor_dim0_stride[31:0]
    g1[6] = (int)((6144u & 0xffffu) << 16);           // d0_stride[47:32]=0 | d1_stride[15:0]
    g1[7] = 0;                                        // tensor_dim1_stride[47:16]
    i32x4 gz = {0, 0, 0, 0};
#if __clang_major__ >= 23
    i32x8 gz8 = {0, 0, 0, 0, 0, 0, 0, 0};
    __builtin_amdgcn_tensor_load_to_lds(g0, g1, gz, gz, gz8, 0);
#else
    __builtin_amdgcn_tensor_load_to_lds(g0, g1, gz, gz, 0);
#endif
  }
#else
  for (int i = tid; i < 3 * LL; i += 256) vlds[i] = vg[i];
#endif

  // K fill (interleaved K-pairs, pad row K=3 with zeros) overlaps the TDM transfer
  for (int l = tid; l < LL; l += 256) {
    klds[0][l][0] = kg[l];            // K row 0
    klds[0][l][1] = kg[LL + l];       // K row 1
    klds[1][l][0] = kg[2 * LL + l];   // K row 2
    klds[1][l][1] = 0.f;              // K row 3 (pad)
  }

  const int lane = tid & 31;
  const int wv   = tid >> 5;
  const int row  = lane & 15;
  const int hi   = lane >> 4;        // which K-pair half this lane feeds
  const int n    = n0 + wv * 16 + row;

  // q = Wc @ x[:,n], pre-scaled by 1/sqrt(MID)  (also overlaps the TDM transfer)
  float xc[C6];
  #pragma unroll
  for (int c = 0; c < C6; ++c) xc[c] = x[(b * C6 + c) * NPT + n];
  const float inv_s = 0.57735026918962576f;
  float q0 = 0.f, q1 = 0.f, q2 = 0.f;
  #pragma unroll
  for (int c = 0; c < C6; ++c) {
    q0 += Wc[0 * C6 + c] * xc[c];
    q1 += Wc[1 * C6 + c] * xc[c];
    q2 += Wc[2 * C6 + c] * xc[c];
  }
  q0 *= inv_s; q1 *= inv_s; q2 *= inv_s;

#ifdef HAVE_TDM
  if (tid < 32) __builtin_amdgcn_s_wait_tensorcnt(0);
#endif
  __syncthreads();

  // WMMA A operand (16x4 f32): lanes 0-15 -> (K0,K1), lanes 16-31 -> (K2,K3=0)
  v2f a;
  a.x = (lane < 16) ? q0 : q2;
  a.y = (lane < 16) ? q1 : 0.f;

  // ---- pass 1: row max over all 2048 scores (software-pipelined B loads) ----
  float mx[8];
  #pragma unroll
  for (int v = 0; v < 8; ++v) mx[v] = -3.4e38f;
  v2f bo; bo.x = klds[hi][row][0]; bo.y = klds[hi][row][1];
  for (int t = 0; t < LL / 16; ++t) {
    int ln = (t + 1) * 16 + row;               // in-bounds thanks to LPAD
    v2f bn; bn.x = klds[hi][ln][0]; bn.y = klds[hi][ln][1];
    v8f acc = {0.f, 0.f, 0.f, 0.f, 0.f, 0.f, 0.f, 0.f};
    acc = __builtin_amdgcn_wmma_f32_16x16x4_f32(false, a, false, bo,
                                                (short)0, acc, false, false);
    #pragma unroll
    for (int v = 0; v < 8; ++v) mx[v] = fmaxf(mx[v], acc[v]);
    bo = bn;
  }
  #pragma unroll
  for (int m = 1; m <= 8; m <<= 1)
    #pragma unroll
    for (int v = 0; v < 8; ++v) mx[v] = fmaxf(mx[v], __shfl_xor(mx[v], m, 32));

  // ---- pass 2: exp, denom, att@v folded in (pipelined B + V loads) ----
  float ds[8], f0[8], f1[8], f2[8];
  #pragma unroll
  for (int v = 0; v < 8; ++v) { ds[v] = 0.f; f0[v] = 0.f; f1[v] = 0.f; f2[v] = 0.f; }
  bo.x = klds[hi][row][0]; bo.y = klds[hi][row][1];
  float vv0 = vlds[row * 3 + 0], vv1 = vlds[row * 3 + 1], vv2 = vlds[row * 3 + 2];
  for (int t = 0; t < LL / 16; ++t) {
    int ln = (t + 1) * 16 + row;
    v2f bn; bn.x = klds[hi][ln][0]; bn.y = klds[hi][ln][1];
    float w0 = vlds[ln * 3 + 0], w1 = vlds[ln * 3 + 1], w2 = vlds[ln * 3 + 2];
    v8f acc = {0.f, 0.f, 0.f, 0.f, 0.f, 0.f, 0.f, 0.f};
    acc = __builtin_amdgcn_wmma_f32_16x16x4_f32(false, a, false, bo,
                                                (short)0, acc, false, false);
    #pragma unroll
    for (int v = 0; v < 8; ++v) {
      float e = __expf(acc[v] - mx[v]);
      ds[v] += e; f0[v] += e * vv0; f1[v] += e * vv1; f2[v] += e * vv2;
    }
    bo = bn; vv0 = w0; vv1 = w1; vv2 = w2;
  }
  #pragma unroll
  for (int m = 1; m <= 8; m <<= 1)
    #pragma unroll
    for (int v = 0; v < 8; ++v) {
      ds[v] += __shfl_xor(ds[v], m, 32);
      f0[v] += __shfl_xor(f0[v], m, 32);
      f1[v] += __shfl_xor(f1[v], m, 32);
      f2[v] += __shfl_xor(f2[v], m, 32);
    }
  if (lane == 0 || lane == 16) {
    int rb = hi * 8;    // lanes 0-15 hold rows 0-7, lanes 16-31 rows 8-15
    #pragma unroll
    for (int v = 0; v < 8; ++v) {
      xch[wv][rb + v][0] = f0[v]; xch[wv][rb + v][1] = f1[v];
      xch[wv][rb + v][2] = f2[v]; xch[wv][rb + v][3] = ds[v];
    }
  }
  __syncthreads();

  // ---- per-point epilogue (lanes>=16 duplicate lane-16's point, masked out) ----
  float inv = 1.f / xch[wv][row][3];
  float fia0 = xch[wv][row][0] * inv;
  float fia1 = xch[wv][row][1] * inv;
  float fia2 = xch[wv][row][2] * inv;

  const float* kin = ws + WS_KIN + b * 9;
  const float* vin = ws + WS_VIN + b * 9;
  float s0 = q0 * kin[0] + q1 * kin[3] + q2 * kin[6];
  float s1 = q0 * kin[1] + q1 * kin[4] + q2 * kin[7];
  float s2 = q0 * kin[2] + q1 * kin[5] + q2 * kin[8];
  float m3 = fmaxf(s0, fmaxf(s1, s2));
  float e0 = __expf(s0 - m3), e1 = __expf(s1 - m3), e2 = __expf(s2 - m3);
  float id = 1.f / (e0 + e1 + e2);
  e0 *= id; e1 *= id; e2 *= id;
  float cf[C6];
  cf[0] = e0 * vin[0] + e1 * vin[3] + e2 * vin[6];
  cf[1] = e0 * vin[1] + e1 * vin[4] + e2 * vin[7];
  cf[2] = e0 * vin[2] + e1 * vin[5] + e2 * vin[8];
  cf[3] = fia0; cf[4] = fia1; cf[5] = fia2;

  // LayerNorm over 6
  float mu = 0.f;
  #pragma unroll
  for (int c = 0; c < C6; ++c) mu += cf[c];
  mu *= (1.f / 6.f);
  float var = 0.f;
  #pragma unroll
  for (int c = 0; c < C6; ++c) { float d = cf[c] - mu; var += d * d; }
  var *= (1.f / 6.f);
  float nr = rsqrtf(var + 1e-5f);
  #pragma unroll
  for (int c = 0; c < C6; ++c) cf[c] = (cf[c] - mu) * nr * lng[c] + lnb[c];

  // y = Wd @ cf   (pre-BN), stored into d_out as scratch
  float y[C6];
  #pragma unroll
  for (int o = 0; o < C6; ++o) {
    float s = 0.f;
    #pragma unroll
    for (int c = 0; c < C6; ++c) s += Wd[o * C6 + c] * cf[c];
    y[o] = s;
  }
  if (lane < 16)
    #pragma unroll
    for (int o = 0; o < C6; ++o) yout[(b * C6 + o) * NPT + n] = y[o];

  // deterministic BN partials: wave reduce -> block reduce -> ws
  float msk = (lane < 16) ? 1.f : 0.f;
  float s1b[C6], s2b[C6];
  #pragma unroll
  for (int o = 0; o < C6; ++o) { s1b[o] = y[o] * msk; s2b[o] = y[o] * y[o] * msk; }
  #pragma unroll
  for (int m = 1; m <= 16; m <<= 1)
    #pragma unroll
    for (int o = 0; o < C6; ++o) {
      s1b[o] += __shfl_xor(s1b[o], m, 32);
      s2b[o] += __shfl_xor(s2b[o], m, 32);
    }
  if (lane == 0)
    #pragma unroll
    for (int o = 0; o < C6; ++o) { bred[wv][o] = s1b[o]; bred[wv][6 + o] = s2b[o]; }
  __syncthreads();
  if (tid < 12) {
    float s = 0.f;
    #pragma unroll
    for (int w = 0; w < 8; ++w) s += bred[w][tid];
    ws[WS_BNP + blockIdx.x * 12 + tid] = s;
  }
}

// ============ Kernel 3: fixed-order BN reduction ============
__global__ void bnred_kernel(float* __restrict__ ws) {
  int t = threadIdx.x;
  if (t < 12) {
    float s = 0.f;
    for (int i = 0; i < NBLK2; ++i) s += ws[WS_BNP + i * 12 + t];
    ws[WS_BNS + t] = s;
  }
}

// ============ Kernel 4: BN + residual + LeakyReLU ============
__global__ __launch_bounds__(256) void finish_kernel(
    const float* __restrict__ x, const float* __restrict__ gamma,
    const float* __restrict__ beta, const float* __restrict__ ws,
    float* __restrict__ out)
{
  int idx = blockIdx.x * 256 + threadIdx.x;
  if (idx >= B2 * C6 * NPT) return;
  int c = (idx / NPT) % C6;
  const float invcnt = 1.f / (float)(B2 * NPT);
  float mean = ws[WS_BNS + c] * invcnt;
  float var  = ws[WS_BNS + 6 + c] * invcnt - mean * mean;
  float rs = rsqrtf(var + 1e-5f);
  float y = out[idx];
  float o = x[idx] + (y - mean) * rs * gamma[c] + beta[c];
  out[idx] = (o >= 0.f) ? o : 0.2f * o;
}

extern "C" void kernel_launch(void* const* d_in, const int* in_sizes, int n_in,
                              void* d_out, int out_size, void* d_ws, size_t ws_size,
                              hipStream_t stream) {
  (void)in_sizes; (void)n_in; (void)out_size; (void)ws_size;
  const float* x      = (const float*)d_in[0];
  const float* curves = (const float*)d_in[1];
  const float* Wa     = (const float*)d_in[2];
  const float* Wav    = (const float*)d_in[3];
  const float* Wb     = (const float*)d_in[4];
  const float* Wbv    = (const float*)d_in[5];
  const float* Wc     = (const float*)d_in[6];
  const float* Wd     = (const float*)d_in[7];
  const float* bng    = (const float*)d_in[8];
  const float* bnb    = (const float*)d_in[9];
  const float* Watt   = (const float*)d_in[10];
  const float* lng    = (const float*)d_in[11];
  const float* lnb    = (const float*)d_in[12];
  const float* Wpl    = (const float*)d_in[13];
  const float* bpl    = (const float*)d_in[14];
  const float* Wpn    = (const float*)d_in[15];
  const float* bpn    = (const float*)d_in[16];
  float* out = (float*)d_out;
  float* ws  = (float*)d_ws;

  prep_kernel<<<B2, 256, 0, stream>>>(curves, Wa, Wav, Wb, Wbv, Watt,
                                      Wpl, bpl, Wpn, bpn, ws);
  attn_kernel<<<NBLK2, 256, 0, stream>>>(x, Wc, Wd, lng, lnb, ws, out);
  bnred_kernel<<<1, 64, 0, stream>>>(ws);
  finish_kernel<<<(B2 * C6 * NPT + 255) / 256, 256, 0, stream>>>(x, bng, bnb, ws, out);
}